// ReconstructCapsNet_58849641890004
// MI455X (gfx1250) — compile-verified
//
#include <hip/hip_runtime.h>
#include <hip/hip_bf16.h>

typedef __attribute__((ext_vector_type(2))) float v2f;
typedef __attribute__((ext_vector_type(8))) float v8f;

#define BSZ    4096
#define INDIM  100
#define DDIM   164
#define DPAIR  (DDIM / 2)
#define MCAP   33
#define NCAP   16
#define TDIM   16
#define CHUNK  512
#define NCHUNK (BSZ / CHUNK)
#define EPSV   1e-5f
#define NEGBIG -3.402823466e38f

__device__ __forceinline__ float sigmoidf(float x) { return 1.0f / (1.0f + expf(-x)); }

// ---------------------------------------------------------------------------
// leaves row-normalize: lh[l][t] = leaves[l][t] / max(||leaves[l]||, 1e-12)
// ---------------------------------------------------------------------------
__global__ void lh_kernel(const float* __restrict__ leaves, float* __restrict__ lh) {
    int t = threadIdx.x;
    if (t < 32) {
        float ss = 0.0f;
        for (int k = 0; k < 16; ++k) { float v = leaves[t * 16 + k]; ss += v * v; }
        float inv = 1.0f / fmaxf(sqrtf(ss), 1e-12f);
        for (int k = 0; k < 16; ++k) lh[t * 16 + k] = leaves[t * 16 + k] * inv;
    }
}

// ---------------------------------------------------------------------------
// Exact entmax-1.5 over one column of length d (d <= 256), 256-thread block.
// Sort-based threshold (bitonic sort + inclusive scans in LDS).
// ---------------------------------------------------------------------------
__device__ float entmax15_block(float v_in, int d,
                                float* s_a, float* s_b, float* s_r) {
    const int tid = threadIdx.x;
    const bool valid = (tid < d);
    // max over valid entries
    s_r[tid] = valid ? v_in : NEGBIG;
    __syncthreads();
    for (int s = 128; s > 0; s >>= 1) {
        if (tid < s) s_r[tid] = fmaxf(s_r[tid], s_r[tid + s]);
        __syncthreads();
    }
    float mx = s_r[0];
    __syncthreads();
    float x = valid ? (v_in - mx) * 0.5f : NEGBIG;
    // bitonic sort descending
    s_a[tid] = x;
    __syncthreads();
    for (int k = 2; k <= 256; k <<= 1) {
        for (int j = k >> 1; j > 0; j >>= 1) {
            int ixj = tid ^ j;
            if (ixj > tid) {
                float a = s_a[tid], b = s_a[ixj];
                bool desc = ((tid & k) == 0);
                if (desc ? (a < b) : (a > b)) { s_a[tid] = b; s_a[ixj] = a; }
            }
            __syncthreads();
        }
    }
    float xs = s_a[tid];
    // inclusive scans of xs and xs^2
    s_b[tid] = xs;
    s_r[tid] = xs * xs;
    __syncthreads();
    for (int off = 1; off < 256; off <<= 1) {
        float av = (tid >= off) ? s_b[tid - off] : 0.0f;
        float aq = (tid >= off) ? s_r[tid - off] : 0.0f;
        __syncthreads();
        s_b[tid] += av;
        s_r[tid] += aq;
        __syncthreads();
    }
    float tau = NEGBIG, flag = 0.0f;
    if (valid) {
        float rho  = (float)(tid + 1);
        float mean = s_b[tid] / rho;
        float msq  = s_r[tid] / rho;
        float ssv  = rho * (msq - mean * mean);
        float dlt  = fmaxf((1.0f - ssv) / rho, 0.0f);
        tau  = mean - sqrtf(dlt);
        flag = (tau <= xs) ? 1.0f : 0.0f;
    }
    __syncthreads();
    s_b[tid] = tau;
    s_r[tid] = flag;
    __syncthreads();
    for (int s = 128; s > 0; s >>= 1) {
        if (tid < s) s_r[tid] += s_r[tid + s];
        __syncthreads();
    }
    int support = (int)(s_r[0] + 0.5f);
    float tau_star = s_b[support - 1];
    float r = fmaxf(x - tau_star, 0.0f);
    return valid ? r * r : 0.0f;
}

// route_w [33][16][164][16], entmax over D.
// Output layout pairs consecutive d rows per column so the GEMM's B fragment
// (rows k0, k0+1 at one column) is a single aligned 8-byte load:
//   wmask[((m*82 + d/2)*256 + n*16+t)*2 + (d&1)]
__global__ __launch_bounds__(256) void entmax_route_kernel(const float* __restrict__ rw,
                                                           float* __restrict__ wmask) {
    __shared__ float s_a[256], s_b[256], s_r[256];
    const int c = blockIdx.x;            // 0..8447
    const int m = c >> 8, n = (c >> 4) & 15, t = c & 15;
    const int tid = threadIdx.x;
    float v = 0.0f;
    if (tid < DDIM) v = rw[((size_t)(m * 16 + n) * DDIM + tid) * 16 + t];
    float o = entmax15_block(v, DDIM, s_a, s_b, s_r);
    if (tid < DDIM) {
        size_t idx = (((size_t)m * DPAIR + (tid >> 1)) * 256 + n * 16 + t) * 2 + (tid & 1);
        wmask[idx] = o;
    }
}

// generic row-wise entmax (decoder masks)
__global__ __launch_bounds__(256) void entmax_rows_kernel(const float* __restrict__ in,
                                                          float* __restrict__ out, int d) {
    __shared__ float s_a[256], s_b[256], s_r[256];
    const int p = blockIdx.x;
    const int tid = threadIdx.x;
    float v = (tid < d) ? in[(size_t)p * d + tid] : 0.0f;
    float o = entmax15_block(v, d, s_a, s_b, s_r);
    if (tid < d) out[(size_t)p * d + tid] = o;
}

// ---------------------------------------------------------------------------
// Encoder: init linear, f = [x, init_x], per-(b,m) LN stats of f*prim_fc[:,m]
// ---------------------------------------------------------------------------
__global__ __launch_bounds__(256) void encoder_kernel(const float* __restrict__ x,
                                                      const float* __restrict__ init_w,
                                                      const float* __restrict__ init_b,
                                                      const float* __restrict__ prim_fc,
                                                      float* __restrict__ f_ws,
                                                      float* __restrict__ mstats) {
    __shared__ float fL[DDIM];
    const int tid = threadIdx.x;
    const size_t b = blockIdx.x;
    if (tid < INDIM) fL[tid] = x[b * INDIM + tid];
    __syncthreads();
    if (tid < 64) {
        float acc = init_b[tid];
        for (int j = 0; j < INDIM; ++j) acc += fL[j] * init_w[tid * INDIM + j];
        fL[INDIM + tid] = acc;
    }
    __syncthreads();
    if (tid < DDIM) f_ws[b * DDIM + tid] = fL[tid];
    const int wid = tid >> 5, lane = tid & 31;
    for (int m = wid; m < MCAP; m += 8) {
        float s = 0.0f, ss = 0.0f;
        for (int d = lane; d < DDIM; d += 32) {
            float pm = (m < 32) ? prim_fc[d * 32 + m] : 1.0f;
            float v = fL[d] * pm;
            s += v; ss += v * v;
        }
        for (int o = 16; o > 0; o >>= 1) {
            s += __shfl_xor(s, o, 32);
            ss += __shfl_xor(ss, o, 32);
        }
        if (lane == 0) {
            float mu = s * (1.0f / (float)DDIM);
            float var = ss * (1.0f / (float)DDIM) - mu * mu;
            mstats[(b * MCAP + m) * 2 + 0] = mu;
            mstats[(b * MCAP + m) * 2 + 1] = rsqrtf(fmaxf(var, 0.0f) + EPSV);
        }
    }
}

// ---------------------------------------------------------------------------
// priors GEMM with V_WMMA_F32_16X16X4_F32:
// per block: one 16-row b-tile x one m; A (16x164 LN'd caps) built in LDS,
// B panel (paired-row layout -> one b64 load per fragment) streamed from L2.
// ---------------------------------------------------------------------------
__global__ __launch_bounds__(256) void caps_gemm_kernel(const float* __restrict__ f_ws,
                                                        const float* __restrict__ mstats,
                                                        const float* __restrict__ prim_fc,
                                                        const float* __restrict__ encg,
                                                        const float* __restrict__ encb,
                                                        const float* __restrict__ wmask,
                                                        float* __restrict__ priors,
                                                        int chunk) {
    __shared__ float As[16 * DDIM];
    __shared__ float s_mu[16], s_isd[16];
    const int tid = threadIdx.x;
    const int m = blockIdx.y;
    const int b0 = chunk * CHUNK + blockIdx.x * 16;
    if (tid < 16) {
        s_mu[tid]  = mstats[((size_t)(b0 + tid) * MCAP + m) * 2 + 0];
        s_isd[tid] = mstats[((size_t)(b0 + tid) * MCAP + m) * 2 + 1];
    }
    __syncthreads();
    for (int idx = tid; idx < 16 * DDIM; idx += 256) {
        int r = idx / DDIM, dd = idx - r * DDIM;
        float fv = f_ws[(size_t)(b0 + r) * DDIM + dd];
        float pm = (m < 32) ? prim_fc[dd * 32 + m] : 1.0f;
        As[idx] = (fv * pm - s_mu[r]) * s_isd[r] * encg[dd] + encb[dd];
    }
    __syncthreads();
    const int wid = tid >> 5, lane = tid & 31;
    const int hf = lane >> 4, ln = lane & 15;   // lane-half selects K pair / row-half
    const float* Wm = wmask + (size_t)m * DPAIR * 512;   // [82][256][2]
    for (int rep = 0; rep < 2; ++rep) {
        const int ntile = wid + rep * 8;
        const int col = ntile * 16 + ln;
        v8f acc = {0.f, 0.f, 0.f, 0.f, 0.f, 0.f, 0.f, 0.f};
        for (int kk = 0; kk < DDIM; kk += 4) {
            int k0 = kk + 2 * hf;
            v2f a;
            a.x = As[ln * DDIM + k0];
            a.y = As[ln * DDIM + k0 + 1];
            v2f bfr = *reinterpret_cast<const v2f*>(
                Wm + (((size_t)(k0 >> 1)) * 256 + col) * 2);
            acc = __builtin_amdgcn_wmma_f32_16x16x4_f32(false, a, false, bfr,
                                                        (short)0, acc, false, false);
        }
#pragma unroll
        for (int j = 0; j < 8; ++j) {
            int row = j + 8 * hf;   // C/D layout: VGPR j -> M=j (lanes 0-15), M=8+j (lanes 16-31)
            priors[((size_t)(blockIdx.x * 16 + row) * MCAP + m) * 256 + col] = acc[j];
        }
    }
}

// ---------------------------------------------------------------------------
// Routing: votes/dis/softmax/hidden-LN/pred/hsel, one block per sample.
// ---------------------------------------------------------------------------
__global__ __launch_bounds__(256) void routing_kernel(const float* __restrict__ priors,
                                                      const float* __restrict__ lh,
                                                      const float* __restrict__ thr,
                                                      const float* __restrict__ capsg,
                                                      const float* __restrict__ capsb,
                                                      const float* __restrict__ y,
                                                      float* __restrict__ pred_out,
                                                      float* __restrict__ hsel_ws,
                                                      int chunk) {
    __shared__ float s_pri[MCAP * 256];
    __shared__ float s_lh[512];
    __shared__ float s_vm[512];
    __shared__ float s_dis[MCAP * 16];
    __shared__ float s_prob[MCAP * 16];
    __shared__ float s_hid[256];
    __shared__ float s_mu[16], s_isd[16], s_nrm[16];
    const int tid = threadIdx.x;
    const int bl = blockIdx.x;
    const size_t b = (size_t)chunk * CHUNK + bl;
    for (int i = tid; i < MCAP * 256; i += 256) s_pri[i] = priors[(size_t)bl * (MCAP * 256) + i];
    for (int i = tid; i < 512; i += 256) s_lh[i] = lh[i];
    __syncthreads();
    // mean-over-m of sigmoid votes
    for (int e = tid; e < 512; e += 256) {
        int n = e >> 5, l = e & 31;
        float acc = 0.0f;
        for (int m = 0; m < MCAP; ++m) {
            const float* pr = &s_pri[(m * 16 + n) * 16];
            const float* lr = &s_lh[l * 16];
            float dot = 0.0f;
            for (int t = 0; t < 16; ++t) dot += lr[t] * pr[t];
            acc += sigmoidf(dot);
        }
        s_vm[e] = acc * (1.0f / 33.0f);
    }
    __syncthreads();
    // dis[m][n] = mean_l (vote - vm)^2 (votes recomputed -> no 67KB LDS buffer)
    for (int me = tid; me < MCAP * 16; me += 256) {
        int m = me >> 4, n = me & 15;
        const float* pr = &s_pri[(m * 16 + n) * 16];
        float acc = 0.0f;
        for (int l = 0; l < 32; ++l) {
            const float* lr = &s_lh[l * 16];
            float dot = 0.0f;
            for (int t = 0; t < 16; ++t) dot += lr[t] * pr[t];
            float dv = sigmoidf(dot) - s_vm[n * 32 + l];
            acc += dv * dv;
        }
        s_dis[me] = acc * (1.0f / 32.0f);
    }
    __syncthreads();
    // softmax over M per n of relu(thread^2 - dis)
    if (tid < 16) {
        int n = tid;
        float mx = NEGBIG;
        for (int m = 0; m < MCAP; ++m) {
            float th = thr[m * 16 + n];
            float w = fmaxf(th * th - s_dis[m * 16 + n], 0.0f);
            mx = fmaxf(mx, w);
        }
        float s = 0.0f;
        for (int m = 0; m < MCAP; ++m) {
            float th = thr[m * 16 + n];
            float w = fmaxf(th * th - s_dis[m * 16 + n], 0.0f);
            s += expf(w - mx);
        }
        float inv = 1.0f / s;
        for (int m = 0; m < MCAP; ++m) {
            float th = thr[m * 16 + n];
            float w = fmaxf(th * th - s_dis[m * 16 + n], 0.0f);
            s_prob[m * 16 + n] = expf(w - mx) * inv;
        }
    }
    __syncthreads();
    // hidden[n][t] = sum_m prob * priors ; then LN over T
    {
        int n = tid >> 4, t = tid & 15;
        float h = 0.0f;
        for (int m = 0; m < MCAP; ++m) h += s_prob[m * 16 + n] * s_pri[(m * 16 + n) * 16 + t];
        s_hid[tid] = h;
    }
    __syncthreads();
    if (tid < 16) {
        float s = 0.0f, ss = 0.0f;
        for (int t = 0; t < 16; ++t) { float v = s_hid[tid * 16 + t]; s += v; ss += v * v; }
        float mu = s * (1.0f / 16.0f);
        float var = ss * (1.0f / 16.0f) - mu * mu;
        s_mu[tid] = mu;
        s_isd[tid] = rsqrtf(fmaxf(var, 0.0f) + EPSV);
    }
    __syncthreads();
    {
        int n = tid >> 4, t = tid & 15;
        float v = (s_hid[tid] - s_mu[n]) * s_isd[n] * capsg[t] + capsb[t];
        s_hid[tid] = v;   // only own element overwritten; cross reads done above
    }
    __syncthreads();
    if (tid < 16) {
        float ss = 0.0f;
        for (int t = 0; t < 16; ++t) { float v = s_hid[tid * 16 + t]; ss += v * v; }
        s_nrm[tid] = sqrtf(ss);
    }
    __syncthreads();
    if (tid < 4) {   // pred: reshape(B, NUM_CLASS, SUB) sum over SUB
        float p = 0.0f;
        for (int s = 0; s < 4; ++s) p += s_nrm[tid * 4 + s];
        pred_out[b * 4 + tid] = p;
    }
    if (tid < 64) {  // hsel: reshape(B, SUB, NUM_CLASS, T) gated by y
        int s = tid >> 4, t = tid & 15;
        float acc = 0.0f;
        for (int c = 0; c < 4; ++c) acc += s_hid[(s * 4 + c) * 16 + t] * y[b * 4 + c];
        hsel_ws[b * 64 + tid] = acc;
    }
}

// ---------------------------------------------------------------------------
// Decoder
// ---------------------------------------------------------------------------
__global__ __launch_bounds__(256) void dec1_fc_kernel(const float* __restrict__ hsel,
                                                      const float* __restrict__ mask1,
                                                      const float* __restrict__ fc1w,
                                                      const float* __restrict__ fc1b,
                                                      float* __restrict__ h1) {
    int gid = blockIdx.x * 256 + threadIdx.x;
    if (gid >= BSZ * 96) return;
    int b = gid / 96, j = gid - b * 96;
    int p = j >> 5;
    const float* msk = mask1 + p * 64;
    const float* w = fc1w + (size_t)j * 64;       // (p*32+o)*64 == j*64
    const float* hs = hsel + (size_t)b * 64;
    float acc = fc1b[j];
    for (int d = 0; d < 64; ++d) acc += msk[d] * hs[d] * w[d];
    h1[gid] = acc;
}

__global__ void gbn_stats_kernel(const float* __restrict__ h, float* __restrict__ gs, int ncol) {
    int chunk = blockIdx.x, j = threadIdx.x;      // blockDim == ncol
    const float* base = h + (size_t)chunk * 256 * ncol + j;
    float s = 0.0f, ss = 0.0f;
    for (int r = 0; r < 256; ++r) { float v = base[(size_t)r * ncol]; s += v; ss += v * v; }
    float mu = s * (1.0f / 256.0f);
    float var = ss * (1.0f / 256.0f) - mu * mu;
    gs[(chunk * ncol + j) * 2 + 0] = mu;
    gs[(chunk * ncol + j) * 2 + 1] = rsqrtf(fmaxf(var, 0.0f) + EPSV);
}

__global__ __launch_bounds__(256) void dec1_glu_kernel(const float* __restrict__ h1,
                                                       const float* __restrict__ gs1,
                                                       const float* __restrict__ bng,
                                                       const float* __restrict__ bnb,
                                                       float* __restrict__ out1) {
    int b = blockIdx.x * 256 + threadIdx.x;
    if (b >= BSZ) return;
    int chunk = b >> 8;
    const float* hr = h1 + (size_t)b * 96;
    const float* g = gs1 + chunk * 96 * 2;
    float o[16];
    for (int k = 0; k < 16; ++k) o[k] = 0.0f;
    for (int p = 0; p < 3; ++p)
        for (int k = 0; k < 16; ++k) {
            int j1 = p * 32 + k, j2 = j1 + 16;
            float a = (hr[j1] - g[j1 * 2]) * g[j1 * 2 + 1] * bng[j1] + bnb[j1];
            float c = (hr[j2] - g[j2 * 2]) * g[j2 * 2 + 1] * bng[j2] + bnb[j2];
            o[k] += fmaxf(sigmoidf(a) * c, 0.0f);
        }
    for (int k = 0; k < 16; ++k) out1[(size_t)b * 16 + k] = o[k];
}

__global__ __launch_bounds__(256) void dec2_fc_kernel(const float* __restrict__ hsel,
                                                      const float* __restrict__ out1,
                                                      const float* __restrict__ mask2,
                                                      const float* __restrict__ fc2w,
                                                      const float* __restrict__ fc2b,
                                                      float* __restrict__ h2) {
    int gid = blockIdx.x * 256 + threadIdx.x;
    if (gid >= BSZ * 192) return;
    int b = gid / 192, j = gid - b * 192;
    int p = j >> 6;
    const float* msk = mask2 + p * 80;
    const float* w = fc2w + (size_t)j * 80;       // (p*64+o)*80 == j*80
    float acc = fc2b[j];
    for (int d = 0; d < 80; ++d) {
        float xv = (d < 64) ? hsel[(size_t)b * 64 + d] : out1[(size_t)b * 16 + (d - 64)];
        acc += msk[d] * xv * w[d];
    }
    h2[gid] = acc;
}

__global__ __launch_bounds__(256) void dec2_glu_kernel(const float* __restrict__ h2,
                                                       const float* __restrict__ gs2,
                                                       const float* __restrict__ bng,
                                                       const float* __restrict__ bnb,
                                                       float* __restrict__ out2) {
    int b = blockIdx.x * 256 + threadIdx.x;
    if (b >= BSZ) return;
    int chunk = b >> 8;
    const float* hr = h2 + (size_t)b * 192;
    const float* g = gs2 + chunk * 192 * 2;
    float o[32];
    for (int k = 0; k < 32; ++k) o[k] = 0.0f;
    for (int p = 0; p < 3; ++p)
        for (int k = 0; k < 32; ++k) {
            int j1 = p * 64 + k, j2 = j1 + 32;
            float a = (hr[j1] - g[j1 * 2]) * g[j1 * 2 + 1] * bng[j1] + bnb[j1];
            float c = (hr[j2] - g[j2 * 2]) * g[j2 * 2 + 1] * bng[j2] + bnb[j2];
            o[k] += fmaxf(sigmoidf(a) * c, 0.0f);
        }
    for (int k = 0; k < 32; ++k) out2[(size_t)b * 32 + k] = o[k];
}

__global__ __launch_bounds__(256) void rec_kernel(const float* __restrict__ out2,
                                                  const float* __restrict__ dec_w,
                                                  const float* __restrict__ dec_b,
                                                  float* __restrict__ out) {
    int gid = blockIdx.x * 256 + threadIdx.x;
    if (gid >= BSZ * 100) return;
    int b = gid / 100, i = gid - b * 100;
    float acc = dec_b[i];
    for (int k = 0; k < 32; ++k) acc += out2[(size_t)b * 32 + k] * dec_w[i * 32 + k];
    out[(size_t)BSZ * 4 + (size_t)b * 100 + i] = acc;
}

// ---------------------------------------------------------------------------
// Workspace layout (float offsets), ~33 MB total
// ---------------------------------------------------------------------------
constexpr size_t OFF_LH    = 0;
constexpr size_t OFF_WMASK = OFF_LH + 512;
constexpr size_t OFF_MASK1 = OFF_WMASK + (size_t)MCAP * DDIM * 256;
constexpr size_t OFF_MASK2 = OFF_MASK1 + 3 * 64;
constexpr size_t OFF_F     = OFF_MASK2 + 3 * 80;
constexpr size_t OFF_MST   = OFF_F + (size_t)BSZ * DDIM;
constexpr size_t OFF_PRI   = OFF_MST + (size_t)BSZ * MCAP * 2;
constexpr size_t OFF_HSEL  = OFF_PRI + (size_t)CHUNK * MCAP * 256;
constexpr size_t OFF_H1    = OFF_HSEL + (size_t)BSZ * 64;
constexpr size_t OFF_GS1   = OFF_H1 + (size_t)BSZ * 96;
constexpr size_t OFF_OUT1  = OFF_GS1 + 16 * 96 * 2;
constexpr size_t OFF_H2    = OFF_OUT1 + (size_t)BSZ * 16;
constexpr size_t OFF_GS2   = OFF_H2 + (size_t)BSZ * 192;
constexpr size_t OFF_OUT2  = OFF_GS2 + 16 * 192 * 2;

extern "C" void kernel_launch(void* const* d_in, const int* in_sizes, int n_in,
                              void* d_out, int out_size, void* d_ws, size_t ws_size,
                              hipStream_t stream) {
    const float* x        = (const float*)d_in[0];
    const float* y        = (const float*)d_in[1];
    const float* init_w   = (const float*)d_in[2];
    const float* init_b   = (const float*)d_in[3];
    const float* enc_ln_g = (const float*)d_in[4];
    const float* enc_ln_b = (const float*)d_in[5];
    const float* prim_fc  = (const float*)d_in[6];
    const float* route_w  = (const float*)d_in[7];
    const float* thr      = (const float*)d_in[8];
    const float* leaves   = (const float*)d_in[9];
    const float* caps_g   = (const float*)d_in[10];
    const float* caps_b   = (const float*)d_in[11];
    const float* m1_w     = (const float*)d_in[12];
    const float* fc1_w    = (const float*)d_in[13];
    const float* fc1_b    = (const float*)d_in[14];
    const float* bn1_g    = (const float*)d_in[15];
    const float* bn1_b    = (const float*)d_in[16];
    const float* m2_w     = (const float*)d_in[17];
    const float* fc2_w    = (const float*)d_in[18];
    const float* fc2_b    = (const float*)d_in[19];
    const float* bn2_g    = (const float*)d_in[20];
    const float* bn2_b    = (const float*)d_in[21];
    const float* dec_w    = (const float*)d_in[22];
    const float* dec_b    = (const float*)d_in[23];
    float* ws  = (float*)d_ws;
    float* out = (float*)d_out;

    lh_kernel<<<1, 32, 0, stream>>>(leaves, ws + OFF_LH);
    entmax_route_kernel<<<MCAP * NCAP * TDIM, 256, 0, stream>>>(route_w, ws + OFF_WMASK);
    entmax_rows_kernel<<<3, 256, 0, stream>>>(m1_w, ws + OFF_MASK1, 64);
    entmax_rows_kernel<<<3, 256, 0, stream>>>(m2_w, ws + OFF_MASK2, 80);
    encoder_kernel<<<BSZ, 256, 0, stream>>>(x, init_w, init_b, prim_fc,
                                            ws + OFF_F, ws + OFF_MST);
    for (int c = 0; c < NCHUNK; ++c) {
        caps_gemm_kernel<<<dim3(CHUNK / 16, MCAP), 256, 0, stream>>>(
            ws + OFF_F, ws + OFF_MST, prim_fc, enc_ln_g, enc_ln_b,
            ws + OFF_WMASK, ws + OFF_PRI, c);
        routing_kernel<<<CHUNK, 256, 0, stream>>>(
            ws + OFF_PRI, ws + OFF_LH, thr, caps_g, caps_b, y,
            out, ws + OFF_HSEL, c);
    }
    dec1_fc_kernel<<<(BSZ * 96) / 256, 256, 0, stream>>>(ws + OFF_HSEL, ws + OFF_MASK1,
                                                         fc1_w, fc1_b, ws + OFF_H1);
    gbn_stats_kernel<<<16, 96, 0, stream>>>(ws + OFF_H1, ws + OFF_GS1, 96);
    dec1_glu_kernel<<<BSZ / 256, 256, 0, stream>>>(ws + OFF_H1, ws + OFF_GS1,
                                                   bn1_g, bn1_b, ws + OFF_OUT1);
    dec2_fc_kernel<<<(BSZ * 192) / 256, 256, 0, stream>>>(ws + OFF_HSEL, ws + OFF_OUT1,
                                                          ws + OFF_MASK2, fc2_w, fc2_b,
                                                          ws + OFF_H2);
    gbn_stats_kernel<<<16, 192, 0, stream>>>(ws + OFF_H2, ws + OFF_GS2, 192);
    dec2_glu_kernel<<<BSZ / 256, 256, 0, stream>>>(ws + OFF_H2, ws + OFF_GS2,
                                                   bn2_g, bn2_b, ws + OFF_OUT2);
    rec_kernel<<<(BSZ * 100 + 255) / 256, 256, 0, stream>>>(ws + OFF_OUT2, dec_w, dec_b, out);
}